// HungarianLoss_77472620085316
// MI455X (gfx1250) — compile-verified
//
#include <hip/hip_runtime.h>
#include <hip/hip_bf16.h>
#include <math.h>

typedef float v2f __attribute__((ext_vector_type(2)));
typedef float v8f __attribute__((ext_vector_type(8)));

#define WPB 8    // waves (=batches) per 256-thread block
#define NN  32   // N slots / targets per batch (== wave32 width)
#define NT  8    // number of types

// ---------------------------------------------------------------------------
// Fused kernel: one wave32 per batch.
//   lane = element index for loads, = row i for softmax, = column j-1 for JV.
// ---------------------------------------------------------------------------
__global__ __launch_bounds__(256) void hungarian_fused(
    const float* __restrict__ pred_j,
    const float* __restrict__ logits,
    const float* __restrict__ target_j,
    const int*   __restrict__ target_types,
    float* __restrict__ ws, int Bv)
{
    __shared__ float s_cost[WPB][NN * NN];   // cost matrix (row-major)
    __shared__ float s_nlp [WPB][NN * NT];   // -log(softmax + eps)  (WMMA A)
    __shared__ float s_lsm [WPB][NN * NT];   // log_softmax          (loss)
    __shared__ float s_pj  [WPB][NN];
    __shared__ float s_tj  [WPB][NN];
    __shared__ int   s_tt  [WPB][NN];
    __shared__ float s_u   [WPB][NN + 1];    // JV row duals u[0..32]
    __shared__ int   s_col [WPB][NN];        // col_of_row

    const int tid  = threadIdx.x;
    const int w    = tid >> 5;
    const int lane = tid & 31;
    const int b    = blockIdx.x * WPB + w;
    if (b >= Bv) return;                      // wave-uniform; no __syncthreads used

    // ---- per-batch vectors -------------------------------------------------
    const size_t base = (size_t)b * NN + lane;
    const float pjv = pred_j[base];
    const float tjv = target_j[base];
    const int   ttv = target_types[base];
    s_pj[w][lane] = pjv;
    s_tj[w][lane] = tjv;
    s_tt[w][lane] = ttv;

    // ---- softmax / log-softmax for row i = lane ----------------------------
    const float* xr = logits + base * NT;
    float x[NT];
    {
        float4 a0 = *reinterpret_cast<const float4*>(xr);
        float4 a1 = *reinterpret_cast<const float4*>(xr + 4);
        x[0]=a0.x; x[1]=a0.y; x[2]=a0.z; x[3]=a0.w;
        x[4]=a1.x; x[5]=a1.y; x[6]=a1.z; x[7]=a1.w;
    }
    float m = x[0]; int am = 0;               // first-index argmax (numpy rule)
#pragma unroll
    for (int t = 1; t < NT; ++t) if (x[t] > m) { m = x[t]; am = t; }
    float e[NT]; float ssum = 0.f;
#pragma unroll
    for (int t = 0; t < NT; ++t) { e[t] = expf(x[t] - m); ssum += e[t]; }
    const float invs = 1.f / ssum;
    const float logs = logf(ssum);
#pragma unroll
    for (int t = 0; t < NT; ++t) {
        s_nlp[w][lane * NT + t] = -logf(e[t] * invs + 1e-8f);
        s_lsm[w][lane * NT + t] = (x[t] - m) - logs;
    }
    __builtin_amdgcn_wave_barrier();

    // ---- cost matrix: ALPHA*|dj| + BETA * ((-logp) x onehot(tt)) via WMMA --
    const int half = lane >> 4;
    const int lo   = lane & 15;
#if __has_builtin(__builtin_amdgcn_wmma_f32_16x16x4_f32)
#pragma unroll
    for (int it = 0; it < 2; ++it) {
#pragma unroll
        for (int jt = 0; jt < 2; ++jt) {
            v8f c = {0.f, 0.f, 0.f, 0.f, 0.f, 0.f, 0.f, 0.f};
            const int tcol = s_tt[w][jt * 16 + lo];
#pragma unroll
            for (int kc = 0; kc < 2; ++kc) {
                const int kb = kc * 4 + half * 2;     // K = 2*half + vgpr
                v2f a, bb;
                a.x  = s_nlp[w][(it * 16 + lo) * NT + kb];
                a.y  = s_nlp[w][(it * 16 + lo) * NT + kb + 1];
                bb.x = (tcol == kb    ) ? 1.f : 0.f;  // one-hot column of B
                bb.y = (tcol == kb + 1) ? 1.f : 0.f;
                c = __builtin_amdgcn_wmma_f32_16x16x4_f32(
                        false, a, false, bb, (short)0, c, false, false);
            }
#pragma unroll
            for (int k = 0; k < 8; ++k) {             // D: M = 8*half + k
                const int Mi = it * 16 + half * 8 + k;
                const int Nj = jt * 16 + lo;
                s_cost[w][Mi * NN + Nj] =
                    fabsf(s_pj[w][Mi] - s_tj[w][Nj]) + 0.5f * c[k];
            }
        }
    }
#else
    for (int i = 0; i < NN; ++i)
        s_cost[w][i * NN + lane] =
            fabsf(s_pj[w][i] - tjv) + 0.5f * s_nlp[w][i * NT + ttv];
#endif
    __builtin_amdgcn_wave_barrier();

    // ---- Jonker-Volgenant LSA: lane = column j-1 ---------------------------
    float v = 0.f, minv = 0.f;
    int   p = 0, way = 0;                     // p[lane+1], way[lane+1]
    s_u[w][lane] = 0.f;
    if (lane == 0) s_u[w][NN] = 0.f;
    __builtin_amdgcn_wave_barrier();

    for (int irow = 1; irow <= NN; ++irow) {
        const int p0 = irow;                  // p[0] = current row
        int j0 = 0;
        minv = __builtin_inff();
        way  = 0;
        bool used = false;
        for (;;) {
            if (j0 > 0) used = used || (lane == (j0 - 1));   // used[j0]=True
            const int   i0  = (j0 == 0) ? p0 : __shfl(p, j0 - 1, 32);
            const float u0  = s_u[w][i0];
            const float cur = s_cost[w][(i0 - 1) * NN + lane] - u0 - v;
            const bool  freeL = !used;
            if (freeL && (cur < minv)) { minv = cur; way = j0; }
            const float masked = freeL ? minv : __builtin_inff();
            float dmin = masked;
#pragma unroll
            for (int off = 16; off >= 1; off >>= 1)
                dmin = fminf(dmin, __shfl_xor(dmin, off, 32));
            const unsigned long long bal = __ballot(masked == dmin);
            const int j1 = __ffsll(bal);      // lowest lane + 1 == column j
            // u[p[used]] += delta  (includes p[0]=p0; all targets distinct)
            if (lane == 0) s_u[w][p0] += dmin;
            if (used)      s_u[w][p ] += dmin;
            v    -= used  ? dmin : 0.f;       // v[used]  -= delta
            minv -= freeL ? dmin : 0.f;       // minv[~used] -= delta
            j0 = j1;
            const int pj0 = __shfl(p, j0 - 1, 32);
            __builtin_amdgcn_wave_barrier();
            if (pj0 == 0) break;
        }
        // augment along alternating path
        while (j0 != 0) {
            const int jn = __shfl(way, j0 - 1, 32);
            const int pn = (jn == 0) ? p0 : __shfl(p, jn - 1, 32);
            if (lane == j0 - 1) p = pn;
            j0 = jn;
        }
    }

    // ---- invert assignment, matched loss -----------------------------------
    s_col[w][p - 1] = lane;                   // col_of_row[p[j]-1] = j-1
    __builtin_amdgcn_wave_barrier();
    const int ccol = s_col[w][lane];          // column matched to row=lane
    float jl = fabsf(pjv - s_tj[w][ccol]);
    const int ttc = s_tt[w][ccol];
    float tl = -s_lsm[w][lane * NT + ttc];
    float ac = (am == ttc) ? 1.f : 0.f;
#pragma unroll
    for (int off = 16; off >= 1; off >>= 1) {
        jl += __shfl_xor(jl, off, 32);
        tl += __shfl_xor(tl, off, 32);
        ac += __shfl_xor(ac, off, 32);
    }
    if (lane == 0) {
        ws[(size_t)b * 4 + 0] = jl;
        ws[(size_t)b * 4 + 1] = tl;
        ws[(size_t)b * 4 + 2] = ac;
    }
}

// ---------------------------------------------------------------------------
// Deterministic final reduction: fixed summation order every call.
// ---------------------------------------------------------------------------
__global__ __launch_bounds__(256) void finalize_reduce(
    const float* __restrict__ ws, float* __restrict__ out, int Bv)
{
    __shared__ float sj[256], st[256], sa[256];
    const int t = threadIdx.x;
    float j = 0.f, tl = 0.f, a = 0.f;
    for (int b = t; b < Bv; b += 256) {
        j  += ws[(size_t)b * 4 + 0];
        tl += ws[(size_t)b * 4 + 1];
        a  += ws[(size_t)b * 4 + 2];
    }
    sj[t] = j; st[t] = tl; sa[t] = a;
    __syncthreads();
    for (int s = 128; s > 0; s >>= 1) {
        if (t < s) { sj[t] += sj[t+s]; st[t] += st[t+s]; sa[t] += sa[t+s]; }
        __syncthreads();
    }
    if (t == 0) {
        const float tot = (float)Bv * (float)NN;
        out[0] = (1.0f * sj[0] + 0.5f * st[0]) / tot;  // loss
        out[1] = sj[0] / tot;                          // j_mae
        out[2] = sa[0] / tot;                          // type_acc
    }
}

extern "C" void kernel_launch(void* const* d_in, const int* in_sizes, int n_in,
                              void* d_out, int out_size, void* d_ws, size_t ws_size,
                              hipStream_t stream) {
    (void)n_in; (void)out_size; (void)ws_size;
    const float* pred_j       = (const float*)d_in[0];
    const float* logits       = (const float*)d_in[1];
    const float* target_j     = (const float*)d_in[2];
    const int*   target_types = (const int*)d_in[3];
    // d_in[4], d_in[5] (masks) are all-ones and unused by the reference math.
    float* out = (float*)d_out;
    float* ws  = (float*)d_ws;

    const int Bv = in_sizes[0] / NN;          // batches
    dim3 grid((Bv + WPB - 1) / WPB);
    hungarian_fused<<<grid, 256, 0, stream>>>(pred_j, logits, target_j,
                                              target_types, ws, Bv);
    finalize_reduce<<<1, 256, 0, stream>>>(ws, out, Bv);
}